// Quantization_layer_2396591751337
// MI455X (gfx1250) — compile-verified
//
#include <hip/hip_runtime.h>

typedef float v2f __attribute__((ext_vector_type(2)));
typedef float v8f __attribute__((ext_vector_type(8)));

#define BATCH  2
#define NBAND  40
#define CH     64
#define TT     1500
#define NCODE  1024
#define NROW   (BATCH * TT)   // 3000
#define TILE_M 64             // four 16-row WMMA sub-tiles share each B fragment
#define LDS_STRIDE 68         // padded dwords per A row -> conflict-free ds_load_b64

__attribute__((amdgpu_waves_per_eu(4)))
__launch_bounds__(256)
__global__ void vq_argmin_kernel(const float* __restrict__ x,
                                 const float* __restrict__ cb,
                                 float* __restrict__ out)
{
    __shared__ float a_lds[TILE_M * LDS_STRIDE];   // 17408 B
    __shared__ float red_s[8 * TILE_M];
    __shared__ int   red_i[8 * TILE_M];
    __shared__ int   win_idx[TILE_M];

    const int tid   = threadIdx.x;
    const int lane  = tid & 31;
    const int wave  = tid >> 5;
    const int mtile = blockIdx.x;
    const int band  = blockIdx.y;
    const int row0  = mtile * TILE_M;

    // ---- stage x tile (64 rows x 64 ch) into LDS; coalesced over t ----
    #pragma unroll
    for (int it = 0; it < 16; ++it) {
        int e = it * 256 + tid;
        int m = e & 63;          // t within tile (consecutive across lanes)
        int c = e >> 6;          // channel
        int n = row0 + m;
        float v = 0.0f;
        if (n < NROW) {
            int batch = n / TT;
            int t = n - batch * TT;
            v = x[(((size_t)batch * NBAND + band) * CH + c) * TT + t];
        }
        a_lds[m * LDS_STRIDE + c] = v;
    }
    __syncthreads();

    // 32-bit A layout (16x4): lane = M (0..15), VGPR0: K=0 / K=2 (lane halves),
    // VGPR1: K=1 / K=3 -> per k-step a lane reads 2 consecutive channels.
    const int hl = lane >> 4;
    const int ll = lane & 15;
    const float* aptr = &a_lds[ll * LDS_STRIDE + 2 * hl];

    const float* cbb = cb + (size_t)band * NCODE * CH;

    float best[4][8];
    int   bidx[4][8];
    #pragma unroll
    for (int s = 0; s < 4; ++s)
        #pragma unroll
        for (int r = 0; r < 8; ++r) { best[s][r] = 3.0e38f; bidx[s][r] = 0x7fffffff; }

    // ---- B-tile loader (one 16-code N-tile -> 16 v2f fragments) ----
    auto load_btile = [&](int nt, v2f* bf) {
        int code = wave * 128 + nt * 16 + ll;        // B layout: lane%16 = N
        const float* cbrow = cbb + (size_t)code * CH + 2 * hl;
        #pragma unroll
        for (int j = 0; j < 16; ++j)
            bf[j] = *(const v2f*)(cbrow + 4 * j);    // global_load_b64 clause
    };

    // ---- compute on one N-tile: 64 WMMAs + |c|^2 + argmin update ----
    auto process = [&](int nt, const v2f* bf) {
        int code = wave * 128 + nt * 16 + ll;
        float nrm = 0.0f;
        #pragma unroll
        for (int j = 0; j < 16; ++j) {
            nrm = fmaf(bf[j].x, bf[j].x, nrm);
            nrm = fmaf(bf[j].y, bf[j].y, nrm);
        }
        v8f acc[4];
        #pragma unroll
        for (int s = 0; s < 4; ++s) acc[s] = (v8f){};

        // opaque zero: blocks LICM/CSE from hoisting the LDS A-fragment loads
        // out of the N-tile loop (which would blow registers and spill).
        int zoff = 0;
        asm volatile("" : "+v"(zoff));
        const float* ap = aptr + zoff;

        #pragma unroll
        for (int j = 0; j < 16; ++j) {
            #pragma unroll
            for (int s = 0; s < 4; ++s) {
                // ds_load_b64, shared base VGPR + immediate offset
                v2f a = *(const v2f*)(ap + s * 16 * LDS_STRIDE + 4 * j);
                acc[s] = __builtin_amdgcn_wmma_f32_16x16x4_f32(
                             false, a, false, bf[j], (short)0, acc[s], false, false);
            }
        }
        nrm += __shfl_xor(nrm, 16, 32);   // combine complementary channel halves

        // dist (minus per-row const |x|^2) = |c|^2 - 2*dot ; C/D layout:
        // VGPR r holds row M = r + 8*(lane/16), N = lane%16 (same code per lane)
        #pragma unroll
        for (int s = 0; s < 4; ++s) {
            #pragma unroll
            for (int r = 0; r < 8; ++r) {
                float d = fmaf(-2.0f, acc[s][r], nrm);
                bool b = (d < best[s][r]) || (d == best[s][r] && code < bidx[s][r]);
                best[s][r] = b ? d    : best[s][r];
                bidx[s][r] = b ? code : bidx[s][r];
            }
        }
    };

    // ---- 8 N-tiles per wave, double-buffered B stream ----
    v2f bA[16], bB[16];
    load_btile(0, bA);
    #pragma unroll 1
    for (int p = 0; p < 4; ++p) {
        load_btile(2 * p + 1, bB);       // prefetch next tile
        process(2 * p, bA);
        if (p < 3) load_btile(2 * p + 2, bA);
        process(2 * p + 1, bB);
    }

    // ---- reduce over the 16 N-lanes (xor masks stay within 16-lane halves) ----
    #pragma unroll
    for (int off = 1; off < 16; off <<= 1) {
        #pragma unroll
        for (int s = 0; s < 4; ++s) {
            #pragma unroll
            for (int r = 0; r < 8; ++r) {
                float os = __shfl_xor(best[s][r], off, 32);
                int   oi = __shfl_xor(bidx[s][r], off, 32);
                bool b = (os < best[s][r]) || (os == best[s][r] && oi < bidx[s][r]);
                best[s][r] = b ? os : best[s][r];
                bidx[s][r] = b ? oi : bidx[s][r];
            }
        }
    }
    if (ll == 0) {
        #pragma unroll
        for (int s = 0; s < 4; ++s) {
            #pragma unroll
            for (int r = 0; r < 8; ++r) {
                int row = s * 16 + hl * 8 + r;
                red_s[wave * TILE_M + row] = best[s][r];
                red_i[wave * TILE_M + row] = bidx[s][r];
            }
        }
    }
    __syncthreads();

    // ---- cross-wave reduction: 64 lanes, one row each over 8 waves ----
    if (tid < TILE_M) {
        int row = tid;
        float bs = red_s[row];
        int   bi = red_i[row];
        #pragma unroll
        for (int w = 1; w < 8; ++w) {
            float s = red_s[w * TILE_M + row];
            int   i = red_i[w * TILE_M + row];
            bool better = (s < bs) || (s == bs && i < bi);
            bs = better ? s : bs;
            bi = better ? i : bi;
        }
        win_idx[row] = bi;
    }
    __syncthreads();

    // ---- gather winning code vectors, write out (coalesced in t) ----
    {
        int m = tid & 63;
        int n = row0 + m;
        if (n < NROW) {
            int batch = n / TT;
            int t = n - batch * TT;
            const float* src = cbb + (size_t)win_idx[m] * CH;
            float* dst = out + (((size_t)batch * NBAND + band) * CH) * TT + t;
            #pragma unroll
            for (int it = 0; it < 16; ++it) {
                int c = (tid >> 6) + 4 * it;
                dst[(size_t)c * TT] = src[c];
            }
        }
    }
}

extern "C" void kernel_launch(void* const* d_in, const int* in_sizes, int n_in,
                              void* d_out, int out_size, void* d_ws, size_t ws_size,
                              hipStream_t stream) {
    const float* x  = (const float*)d_in[0];   // (2, 40, 64, 1500) fp32
    const float* cb = (const float*)d_in[1];   // (40, 1024, 64) fp32
    float* out = (float*)d_out;                // (2, 40, 64, 1500) fp32
    dim3 grid((NROW + TILE_M - 1) / TILE_M, NBAND);
    vq_argmin_kernel<<<grid, 256, 0, stream>>>(x, cb, out);
}